// DCT_Features_9972914061876
// MI455X (gfx1250) — compile-verified
//
#include <hip/hip_runtime.h>

typedef __attribute__((ext_vector_type(2))) float v2f;
typedef __attribute__((ext_vector_type(8))) float v8f;
typedef __attribute__((ext_vector_type(4))) int v4i;

#define SLOPE 0.001f
#define PI_F 3.14159265358979323846f

// ---- CDNA5 async global->LDS path (guarded; fallback = sync staging) ------
#if __has_builtin(__builtin_amdgcn_global_load_async_to_lds_b128)
#define HAVE_ASYNC 1
typedef __attribute__((address_space(1))) v4i* gp1_t;   // global v4i*
typedef __attribute__((address_space(3))) v4i* lp3_t;   // LDS v4i*
#else
#define HAVE_ASYNC 0
#endif

#if __has_builtin(__builtin_amdgcn_s_wait_asynccnt)
#define WAIT_ASYNC(n) __builtin_amdgcn_s_wait_asynccnt(n)
#else
#define WAIT_ASYNC(n) asm volatile("s_wait_asynccnt " #n)
#endif

// ---------------------------------------------------------------------------
// Prep: fold 3D DCT matrices into conv weights.
// Output layout (K-pair interleaved so GEMM B-frags are single b64 loads):
//   Wt[s][k2=0..2047][col=0..127][p=0..1],  K index = 2*k2 + p
// value = 0.5 * Wfold[s,col,t,h,w] with K = n*2048 + t*64 + h*8 + w (same for
// both dct bins n, since the reference averages them).
// ---------------------------------------------------------------------------
__global__ __launch_bounds__(256)
void dct_fold_weights(const float* __restrict__ conv_w,
                      float* __restrict__ Wt) {
  __shared__ float ct[32 * 32];   // Ct[f][t] = 2 cos(pi (2t+1) f / 64)
  __shared__ float c8[64];        // C8[g][h] = 2 cos(pi (2h+1) g / 16)
  __shared__ float bufA[2048];
  __shared__ float bufB[2048];

  const int tid = threadIdx.x;
  const int s = blockIdx.x >> 7;
  const int k = blockIdx.x & 127;
  const float* w = conv_w + (size_t)(s * 128 + k) * 2048;

  for (int i = tid; i < 2048; i += 256) bufB[i] = w[i];
  for (int i = tid; i < 1024; i += 256) {
    int f = i >> 5, t = i & 31;
    ct[i] = 2.0f * cosf(PI_F * (float)((2 * t + 1) * f) / 64.0f);
  }
  if (tid < 64) {
    int g = tid >> 3, h = tid & 7;
    c8[tid] = 2.0f * cosf(PI_F * (float)((2 * h + 1) * g) / 16.0f);
  }
  __syncthreads();

  // pass 1: contract f -> t
  for (int i = tid; i < 2048; i += 256) {
    int t = i >> 6, gj = i & 63;
    float acc = 0.0f;
#pragma unroll 8
    for (int f = 0; f < 32; ++f) acc += bufB[f * 64 + gj] * ct[f * 32 + t];
    bufA[i] = acc;
  }
  __syncthreads();

  // pass 2: contract g -> h
  for (int i = tid; i < 2048; i += 256) {
    int t = i >> 6, h = (i >> 3) & 7, j = i & 7;
    float acc = 0.0f;
#pragma unroll
    for (int g = 0; g < 8; ++g) acc += bufA[t * 64 + g * 8 + j] * c8[g * 8 + h];
    bufB[i] = acc;
  }
  __syncthreads();

  // pass 3: contract j -> w, scale 0.5, write both dct-bin copies in the
  // K-pair-interleaved layout.
  for (int i = tid; i < 2048; i += 256) {
    int t = i >> 6, h = (i >> 3) & 7, w2 = i & 7;
    float acc = 0.0f;
#pragma unroll
    for (int j = 0; j < 8; ++j) acc += bufB[t * 64 + h * 8 + j] * c8[j * 8 + w2];
    acc *= 0.5f;
    // K index i (bin n=0) and i+2048 (bin n=1); k2 = K>>1, p = K&1.
    size_t o0 = ((size_t)s * 2048 + (size_t)(i >> 1)) * 256 + (size_t)k * 2 + (i & 1);
    Wt[o0] = acc;
    Wt[o0 + (size_t)1024 * 256] = acc;   // (i+2048)>>1 = (i>>1)+1024
  }
}

// ---------------------------------------------------------------------------
// Main GEMM: out[b, s*128+k] = leaky( sum_c x[b, s*4096+c] * W[s][c][k] + b )
// M=4096, N=128, K=4096 per subwindow s. Bandwidth bound (~64 FLOP/B).
// Double-buffered LDS fed by GLOBAL_LOAD_ASYNC_TO_LDS_B128 (ASYNCcnt),
// fp32 V_WMMA_F32_16X16X4_F32 compute overlapped with the HBM stream.
// ---------------------------------------------------------------------------
#define MT 64
#define KC 32
#define PA 36    // A row stride (floats): conflict-free b64 A-frag loads
#define PBF 288  // B pair-row stride (floats = 128 cols * 2 + 32 pad):
                 // lane-half k2 offset lands 32 banks away -> conflict-free

__device__ __forceinline__ void issue_tile_loads(const float* __restrict__ xrow,
                                                 const float* __restrict__ Bsrc,
                                                 int kt, float* As, float* Bs,
                                                 int tid) {
  // A tile: 64 rows x 32 K-floats (512 x b128, 2 per thread), coalesced.
#pragma unroll
  for (int i = 0; i < 2; ++i) {
    int idx = tid + i * 256;
    int row = idx >> 3;
    int c4 = (idx & 7) << 2;
    const float* g = xrow + (size_t)row * 8192 + kt + c4;
    float* l = &As[row * PA + c4];
#if HAVE_ASYNC
    __builtin_amdgcn_global_load_async_to_lds_b128((gp1_t)g, (lp3_t)l, 0, 0);
#else
    *(float4*)l = *(const float4*)g;
#endif
  }
  // B tile: 16 K-pair-rows x 256 floats (1024 x b128, 4 per thread); the
  // global pair-interleaved layout maps 1:1 onto contiguous LDS pair-rows.
#pragma unroll
  for (int i = 0; i < 4; ++i) {
    int idx = tid + i * 256;
    int row = idx >> 6;              // pair-row 0..15 (64 float4 per row)
    int c4 = (idx & 63) << 2;        // float offset within pair-row 0..252
    const float* g = Bsrc + (size_t)((kt >> 1) + row) * 256 + c4;
    float* l = &Bs[row * PBF + c4];
#if HAVE_ASYNC
    __builtin_amdgcn_global_load_async_to_lds_b128((gp1_t)g, (lp3_t)l, 0, 0);
#else
    *(float4*)l = *(const float4*)g;
#endif
  }
}

__global__ __launch_bounds__(256)
void dct_gemm(const float* __restrict__ x, const float* __restrict__ Wt,
              const float* __restrict__ bias, float* __restrict__ out) {
  __shared__ float As[2][MT * PA];          // 2 x 64 x 36   (18 KB)
  __shared__ float Bs[2][(KC / 2) * PBF];   // 2 x 16 x 288  (36 KB)

  const int tid = threadIdx.x;
  const int s = blockIdx.y;
  const int m0 = blockIdx.x * MT;
  const int wv = tid >> 5;
  const int lane = tid & 31;
  const int hi = lane >> 4;       // lane half (ISA A/B K-split, C M-split)
  const int lm = lane & 15;
  const int waveM = wv >> 2;      // 0..1 -> +32 rows
  const int waveN = wv & 3;       // 0..3 -> +32 cols

  const float* xrow = x + (size_t)m0 * 8192 + (size_t)s * 4096;
  const float* Bsrc = Wt + (size_t)s * 2048 * 256;

  v8f acc[2][2];
#pragma unroll
  for (int i = 0; i < 2; ++i)
#pragma unroll
    for (int j = 0; j < 2; ++j) acc[i][j] = {};

  // Prologue: start streaming chunk 0.
  issue_tile_loads(xrow, Bsrc, 0, As[0], Bs[0], tid);

  for (int kt = 0; kt < 4096; kt += KC) {
    const int cur = (kt >> 5) & 1;
    const bool more = (kt + KC) < 4096;
    // Kick off the next chunk before waiting: overlap HBM with WMMA.
    if (more)
      issue_tile_loads(xrow, Bsrc, kt + KC, As[cur ^ 1], Bs[cur ^ 1], tid);
#if HAVE_ASYNC
    // In-order async completion: <=6 outstanding means current chunk landed.
    if (more) WAIT_ASYNC(6);
    else      WAIT_ASYNC(0);
#endif
    __syncthreads();

#pragma unroll
    for (int k0 = 0; k0 < KC; k0 += 4) {
      const int kb = k0 + 2 * hi;       // per-lane-half K base (ISA 16x4)
      const int k2 = kb >> 1;           // pair-row index
      v2f a[2], b[2];
#pragma unroll
      for (int tm = 0; tm < 2; ++tm) {
        int r = waveM * 32 + tm * 16 + lm;   // A: lane = row lm, K=kb,kb+1
        a[tm] = *(const v2f*)(&As[cur][r * PA + kb]);
      }
#pragma unroll
      for (int tn = 0; tn < 2; ++tn) {
        int c = waveN * 32 + tn * 16 + lm;   // B: lane = col c, K=kb,kb+1
        b[tn] = *(const v2f*)(&Bs[cur][k2 * PBF + c * 2]);
      }
#pragma unroll
      for (int tm = 0; tm < 2; ++tm)
#pragma unroll
        for (int tn = 0; tn < 2; ++tn)
          acc[tm][tn] = __builtin_amdgcn_wmma_f32_16x16x4_f32(
              false, a[tm], false, b[tn], (short)0, acc[tm][tn], false, false);
    }
    __syncthreads();   // protect buffer `cur` before it is refilled
  }

  // Epilogue: bias + LeakyReLU, C layout: VGPR r -> M = r + 8*hi, N = lm.
#pragma unroll
  for (int tm = 0; tm < 2; ++tm) {
#pragma unroll
    for (int tn = 0; tn < 2; ++tn) {
      int col = waveN * 32 + tn * 16 + lm;
      float bv = bias[s * 128 + col];
      int gcol = s * 128 + col;
#pragma unroll
      for (int r = 0; r < 8; ++r) {
        int row = m0 + waveM * 32 + tm * 16 + hi * 8 + r;
        float v = acc[tm][tn][r] + bv;
        v = (v >= 0.0f) ? v : SLOPE * v;
        out[(size_t)row * 256 + gcol] = v;
      }
    }
  }
}

// ---------------------------------------------------------------------------
extern "C" void kernel_launch(void* const* d_in, const int* in_sizes, int n_in,
                              void* d_out, int out_size, void* d_ws, size_t ws_size,
                              hipStream_t stream) {
  const float* x = (const float*)d_in[0];        // [4096,1,128,8,8]
  const float* conv_w = (const float*)d_in[1];   // [2,128,32,8,8]
  const float* conv_b = (const float*)d_in[2];   // [2,128]
  float* out = (float*)d_out;                    // [4096,256]
  float* Wt = (float*)d_ws;                      // [2][2048][128][2] = 4 MB

  dct_fold_weights<<<256, 256, 0, stream>>>(conv_w, Wt);
  dim3 grid(4096 / MT, 2);
  dct_gemm<<<grid, 256, 0, stream>>>(x, Wt, conv_b, out);
}